// PositionalBias_80015240724988
// MI455X (gfx1250) — compile-verified
//
#include <hip/hip_runtime.h>

#define BB     4
#define SS     2048
#define HH     12
#define DD     64
#define SEQ    2046          // S - 2
#define CHUNK  64
#define NCHUNK (SS / CHUNK)  // 32
#define VSTR   72            // LDS row stride (floats): 2*72 % 64 = 16 banks
                             // -> both lane halves of a B load hit disjoint banks

typedef float v2f __attribute__((ext_vector_type(2)));
typedef float v8f __attribute__((ext_vector_type(8)));

#if defined(__AMDGCN__) && __has_builtin(__builtin_amdgcn_global_load_async_to_lds_b128) && \
    __has_builtin(__builtin_amdgcn_s_wait_asynccnt)
#define USE_ASYNC 1
// Exact parameter types of the builtin: (v4i AS1*, v4i AS3*, imm, imm)
typedef int b128v __attribute__((vector_size(16)));
typedef __attribute__((address_space(1))) b128v b128_global;
typedef __attribute__((address_space(3))) b128v b128_shared;
#else
#define USE_ASYNC 0
#endif

// Stage one 64-row x 64-col chunk of V[n,:,h,:] into LDS (1024 float4,
// 4 per thread), rows padded to VSTR. Async path: GLOBAL_LOAD_ASYNC_TO_LDS_B128.
__device__ __forceinline__ void stage_chunk(const float* __restrict__ vbase,
                                            int l_base, float* dst, int tid)
{
#pragma unroll
    for (int k = 0; k < 4; ++k) {
        const int i = tid + k * 256;
        const int r = i >> 4;            // row within chunk
        const int q = (i & 15) * 4;      // float offset within row
        const float* src = vbase + (size_t)(l_base + r) * (HH * DD) + q;
        float*       d   = dst + r * VSTR + q;
#if USE_ASYNC
        __builtin_amdgcn_global_load_async_to_lds_b128(
            (b128_global*)(b128v*)(void*)const_cast<float*>(src),
            (b128_shared*)(b128v*)(void*)d,
            0, 0);
#else
        *(float4*)d = *(const float4*)src;
#endif
    }
}

// ---------------------------------------------------------------------------
// pbv[n,j,h,d] = sum_l bias[h,l,j] * v[n,l,h,d]
// Per (n,h): Out(2048x64) = Bias(2048x2048) * V(2048x64), Bias symmetric
// Toeplitz built from w[h,:]. V_WMMA_F32_16X16X4_F32, K-loop of 512 steps.
// Block = 256 threads = 8 waves; wave wv owns 32 j-rows (two 16x64 C tiles)
// so each shared B fragment feeds 2 WMMAs (8 wmma : 6 ds per k-step).
// ---------------------------------------------------------------------------
__global__ __launch_bounds__(256) void
pbv_wmma_kernel(const float* __restrict__ v,
                const float* __restrict__ w,
                float* __restrict__ out)
{
    __shared__ float wsym2[4096];             // 16 KB: wsym2[u] = w[h,|u-2048|]
    __shared__ float vbuf[2][CHUNK * VSTR];   // 2 x 18 KB ping-pong V chunks

    const int h    = blockIdx.y;
    const int n    = blockIdx.z;
    const int jblk = blockIdx.x;              // 0..7 -> j base jblk*256
    const int tid  = (int)threadIdx.x;
    const int wv   = tid >> 5;
    const int lane = tid & 31;
    const int hlf  = lane >> 4;               // lane half -> K pair (A/B layout)
    const int nidx = lane & 15;

    // Symmetric-extended bias row; unconditional clamped load (w rows are
    // 2048 wide so min(dd,2047) never leaves the row), then select zero.
    for (int u = tid; u < 4096; u += 256) {
        int dd = u - 2048; dd = dd < 0 ? -dd : dd;
        const int   idx = dd < 2047 ? dd : 2047;
        const float val = w[h * SS + idx];
        wsym2[u] = (dd < SEQ) ? val : 0.0f;
    }

    const int j0 = jblk * 256 + wv * 32;      // wave's 32-row j base
    // Two A tiles: rows j0+nidx (tile 0) and j0+16+nidx (tile 1).
    const float* pA0 = &wsym2[2048 - (j0 + nidx) + 2 * hlf];
    const float* pA1 = pA0 - 16;

    // v[n,l,h,d] = vbase[l*768 + d]
    const float* vbase = v + (size_t)n * SS * HH * DD + (size_t)h * DD;

    v8f acc0[4] = {};
    v8f acc1[4] = {};

    stage_chunk(vbase, 0, vbuf[0], tid);      // prefetch chunk 0

    for (int c = 0; c < NCHUNK; ++c) {
        const int cur    = c & 1;
        const int l_base = c * CHUNK;

        if (c + 1 < NCHUNK) {
            stage_chunk(vbase, l_base + CHUNK, vbuf[1 - cur], tid); // prefetch
#if USE_ASYNC
            __builtin_amdgcn_s_wait_asynccnt(4);  // chunk c landed; c+1 in flight
#endif
        } else {
#if USE_ASYNC
            __builtin_amdgcn_s_wait_asynccnt(0);
#endif
        }

        // Structural zeros of bias columns l=0 and l=2047: zero staged V rows
        // instead of masking A in the inner loop.
        if (c == 0 && tid < 16) {
            float4 z = {0.f, 0.f, 0.f, 0.f};
            *(float4*)&vbuf[cur][0 * VSTR + tid * 4] = z;
        }
        if (c == NCHUNK - 1 && tid >= 240) {
            float4 z = {0.f, 0.f, 0.f, 0.f};
            *(float4*)&vbuf[cur][63 * VSTR + (tid - 240) * 4] = z;
        }
        __syncthreads();                      // chunk c visible to all 8 waves

        const float* pAc0 = pA0 + l_base;     // A: imm-offset LDS loads
        const float* pAc1 = pA1 + l_base;
        const float* pBc  = &vbuf[cur][2 * hlf * VSTR + nidx];
#pragma unroll
        for (int kk = 0; kk < CHUNK; kk += 4) {
            v2f a0, a1;                       // A 16x4: lanes 0-15 K0,K1; 16-31 K2,K3
            a0.x = pAc0[kk];
            a0.y = pAc0[kk + 1];
            a1.x = pAc1[kk];
            a1.y = pAc1[kk + 1];
            const float* pB = pBc + kk * VSTR;
#pragma unroll
            for (int c4 = 0; c4 < 4; ++c4) {  // B 4x16: VGPR0={K0|K2}, VGPR1={K1|K3}
                v2f b;
                b.x = pB[c4 * 16];
                b.y = pB[c4 * 16 + VSTR];
                acc0[c4] = __builtin_amdgcn_wmma_f32_16x16x4_f32(
                    false, a0, false, b, (short)0, acc0[c4], false, false);
                acc1[c4] = __builtin_amdgcn_wmma_f32_16x16x4_f32(
                    false, a1, false, b, (short)0, acc1[c4], false, false);
            }
        }
        __syncthreads();                      // all waves done reading buf[cur]
    }

    // Writeback; C/D layout: VGPR r -> M=r (lanes 0-15), M=r+8 (lanes 16-31).
    // Output rows j=0 and j=2047 are structurally zero (bias row padding).
    float* obase = out + (size_t)n * SS * HH * DD + (size_t)h * DD;
#pragma unroll
    for (int c4 = 0; c4 < 4; ++c4) {
#pragma unroll
        for (int r = 0; r < 8; ++r) {
            const int j0r = j0 + r + 8 * hlf;
            const int j1r = j0r + 16;
            const float v0 = (j0r == 0) ? 0.0f : acc0[c4][r];
            const float v1 = (j1r == SS - 1) ? 0.0f : acc1[c4][r];
            obase[(size_t)j0r * (HH * DD) + c4 * 16 + nidx] = v0;
            obase[(size_t)j1r * (HH * DD) + c4 * 16 + nidx] = v1;
        }
    }
}

// ---------------------------------------------------------------------------
// z_pb[0, s, h] = sum_{j=1}^{2046} w[h, |s-j|] for s in [1,2046], else 0.
// 24576 outputs, w is 96 KB (L2 resident): brute force, negligible cost.
// ---------------------------------------------------------------------------
__global__ __launch_bounds__(256) void
zpb_kernel(const float* __restrict__ w, float* __restrict__ z)
{
    const int idx = (int)(blockIdx.x * 256 + threadIdx.x);
    if (idx >= SS * HH) return;
    const int s = idx / HH;
    const int h = idx - s * HH;
    float acc = 0.0f;
    if (s >= 1 && s <= SEQ) {
        const float* wr = w + h * SS;
        for (int j = 1; j <= SEQ; ++j) {
            int d = s - j; d = d < 0 ? -d : d;
            acc += wr[d];
        }
    }
    z[idx] = acc;   // z_pb flat: s*H + h
}

extern "C" void kernel_launch(void* const* d_in, const int* in_sizes, int n_in,
                              void* d_out, int out_size, void* d_ws, size_t ws_size,
                              hipStream_t stream)
{
    const float* v = (const float*)d_in[0];   // (4, 2048, 12, 64) f32
    const float* w = (const float*)d_in[1];   // (12, 2048) f32
    float* pbv = (float*)d_out;                         // 6,291,456 f32
    float* zpb = pbv + (size_t)BB * SS * HH * DD;       //    24,576 f32

    dim3 grid(SS / 256, HH, BB);   // (8, 12, 4)
    pbv_wmma_kernel<<<grid, 256, 0, stream>>>(v, w, pbv);

    const int zt = SS * HH;
    zpb_kernel<<<(zt + 255) / 256, 256, 0, stream>>>(w, zpb);
}